// YOLOV3Loss_36447092474372
// MI455X (gfx1250) — compile-verified
//
#include <hip/hip_runtime.h>
#include <cstdint>

typedef __attribute__((ext_vector_type(2))) float v2f;
typedef __attribute__((ext_vector_type(8))) float v8f;

#define BATCH   16
#define NANCH   3
#define GRID    76
#define A_TOTAL (NANCH*GRID*GRID)          // 17328
#define NCLS    80
#define NBOX    32
#define BLK     256
#define NBLK    ((A_TOTAL + BLK - 1)/BLK)  // 68

__device__ __forceinline__ float sigmoidf_(float x) { return 1.0f / (1.0f + expf(-x)); }

// Stage 1: per (image, anchor-block): IoU max/argmax vs 32 boxes (boxes staged in
// LDS via CDNA5 async global->LDS), sparse BCE + regression for positive anchors,
// deterministic block-level tree reduction into per-block partials.
__global__ void yolo_main(const float* __restrict__ pred,   // (B, A, 85)
                          const float* __restrict__ ann,    // (B, 32, 5)
                          const float* __restrict__ anch,   // (A, 4)
                          float* __restrict__ part)         // (B, NBLK, 3)
{
  const int b   = blockIdx.y;
  const int tid = threadIdx.x;
  const int a   = blockIdx.x * BLK + tid;

  __shared__ float rawann[NBOX * 5];
  __shared__ float bx1[NBOX], by1[NBOX], bx2[NBOX], by2[NBOX], barea[NBOX];
  __shared__ float bcx[NBOX], bcy[NBOX], bww[NBOX], bhh[NBOX], bcl[NBOX];

  // ---- CDNA5 async global->LDS load of the annotation tile (ASYNCcnt path) ----
  {
    const unsigned long long gbase = (unsigned long long)(ann + (size_t)b * NBOX * 5);
    if (tid < NBOX * 5) {
      unsigned ldsaddr = (unsigned)(size_t)(&rawann[tid]);  // low 32 bits == LDS offset
      unsigned voff    = (unsigned)(tid * 4);
      asm volatile("global_load_async_to_lds_b32 %0, %1, %2"
                   :: "v"(ldsaddr), "v"(voff), "s"(gbase) : "memory");
    }
    asm volatile("s_wait_asynccnt 0" ::: "memory");
    __syncthreads();
  }

  // top-left -> center conversion (threads 0..31, one box each)
  if (tid < NBOX) {
    float x = rawann[tid*5+0], y = rawann[tid*5+1];
    float w = rawann[tid*5+2], h = rawann[tid*5+3];
    float cl = rawann[tid*5+4];
    float cx = x + 0.5f*w, cy = y + 0.5f*h;
    bcx[tid]=cx; bcy[tid]=cy; bww[tid]=w; bhh[tid]=h; bcl[tid]=cl;
    bx1[tid]=cx-0.5f*w; by1[tid]=cy-0.5f*h;
    bx2[tid]=cx+0.5f*w; by2[tid]=cy+0.5f*h;
    barea[tid]=w*h;
  }
  __syncthreads();

  float c_acc = 0.f, r_acc = 0.f, n_acc = 0.f;
  if (a < A_TOTAL) {
    const float4 av = ((const float4*)anch)[a];
    const float acx=av.x, acy=av.y, aw=av.z, ah=av.w;
    const float ax1=acx-0.5f*aw, ay1=acy-0.5f*ah;
    const float ax2=acx+0.5f*aw, ay2=acy+0.5f*ah;
    const float areaA = aw*ah;
    float best = -__builtin_inff();
    int bi = 0;
    #pragma unroll 8
    for (int m = 0; m < NBOX; ++m) {
      float iw = fminf(ax2, bx2[m]) - fmaxf(ax1, bx1[m]);
      float ih = fminf(ay2, by2[m]) - fmaxf(ay1, by1[m]);
      iw = fmaxf(iw, 0.f); ih = fmaxf(ih, 0.f);
      float inter = iw * ih;
      float iou = inter / (areaA + barea[m] - inter);
      iou = (bcl[m] != -1.0f) ? iou : -__builtin_inff();  // padded boxes never win
      if (iou > best) { best = iou; bi = m; }             // first-occurrence argmax
    }
    if (best >= 0.5f) {                                   // positive anchor: sparse reads
      n_acc = 1.f;
      const float* pr = pred + ((size_t)b * A_TOTAL + a) * (NCLS + 5);
      float gw = fmaxf(bww[bi], 1.f), gh = fmaxf(bhh[bi], 1.f);
      float stx = sigmoidf_(bcx[bi] - acx);
      float sty = sigmoidf_(bcy[bi] - acy);
      float twg = logf(gw / aw), thg = logf(gh / ah);
      float d0 = sigmoidf_(pr[0]) - stx;
      float d1 = sigmoidf_(pr[1]) - sty;
      float d2 = pr[2] - twg;
      float d3 = pr[3] - thg;
      float d4 = pr[4] - 1.0f;
      r_acc = d0*d0 + d1*d1 + d2*d2 + d3*d3 + d4*d4;
      int k = (int)bcl[bi];
      float s = 0.f;
      for (int c = 0; c < NCLS; ++c) {
        float p = pr[5 + c];
        s -= (c == k) ? logf(p) : logf(1.0f - p);         // BCE with one-hot target
      }
      c_acc = s;
    }
  }

  // deterministic block tree reduction
  __shared__ float rc[BLK], rr[BLK], rn[BLK];
  rc[tid] = c_acc; rr[tid] = r_acc; rn[tid] = n_acc;
  __syncthreads();
  for (int st = BLK/2; st > 0; st >>= 1) {
    if (tid < st) { rc[tid]+=rc[tid+st]; rr[tid]+=rr[tid+st]; rn[tid]+=rn[tid+st]; }
    __syncthreads();
  }
  if (tid == 0) {
    float* o = part + ((size_t)b * NBLK + blockIdx.x) * 3;
    o[0] = rc[0]; o[1] = rr[0]; o[2] = rn[0];
  }
}

// Stage 2 (1 wave): sequential (deterministic) per-image sums over block partials,
// per-image losses, then batch mean via V_WMMA_F32_16X16X4_F32 row-sums:
// rows 0-3 of A hold the 16 cls losses, rows 4-7 hold the 16 reg losses; B = ones.
__global__ void yolo_final(const float* __restrict__ part, float* __restrict__ out)
{
  const int t = threadIdx.x;
  __shared__ float z[32];
  float cls_l = 0.f, reg_l = 0.f;
  if (t < BATCH) {
    float cs = 0.f, rs = 0.f, ns = 0.f;
    for (int i = 0; i < NBLK; ++i) {
      const float* p = part + ((size_t)t * NBLK + i) * 3;
      cs += p[0]; rs += p[1]; ns += p[2];
    }
    float denom = fmaxf(ns, 1.f);
    cls_l = cs / denom;            // (bce*posf).sum() / denom
    reg_l = rs / (denom * denom);  // (Σ 5 masked-mse sums)/denom /denom
  }
  if (t < 16) { z[t] = cls_l; z[16 + t] = reg_l; }
  __syncthreads();

  // A[m][k] = z[4m+k] for m<8, else 0.  Lane L<16: (M=L, K=0/1); L>=16: (M=L-16, K=2/3)
  int m  = t & 15;
  int kb = (t < 16) ? 0 : 2;
  float a0 = (m < 8) ? z[4*m + kb]     : 0.f;
  float a1 = (m < 8) ? z[4*m + kb + 1] : 0.f;
  v2f Am = {a0, a1};
  v2f Bm = {1.f, 1.f};              // all-ones B: layout-independent
  v8f Cm = {0.f,0.f,0.f,0.f,0.f,0.f,0.f,0.f};
  v8f Dm = __builtin_amdgcn_wmma_f32_16x16x4_f32(false, Am, false, Bm,
                                                 (short)0, Cm, false, false);
  // lane 0 holds (M=0..7, N=0) in D[0..7]
  if (t == 0) {
    out[0] = (Dm[0] + Dm[1] + Dm[2] + Dm[3]) * (1.0f / 16.0f);  // cls mean
    out[1] = (Dm[4] + Dm[5] + Dm[6] + Dm[7]) * (1.0f / 16.0f);  // reg mean
  }
}

extern "C" void kernel_launch(void* const* d_in, const int* in_sizes, int n_in,
                              void* d_out, int out_size, void* d_ws, size_t ws_size,
                              hipStream_t stream) {
  const float* pred = (const float*)d_in[0];  // predictions (16,3,76,76,85) f32
  const float* ann  = (const float*)d_in[1];  // bbox_annotations (16,32,5) f32
  const float* anch = (const float*)d_in[2];  // anchors (3,76,76,4) f32
  float* out  = (float*)d_out;                // [cls_loss, reg_loss]
  float* part = (float*)d_ws;                 // (16, 68, 3) f32 partials, ~13 KB

  dim3 grid(NBLK, BATCH);
  hipLaunchKernelGGL(yolo_main, grid, dim3(BLK), 0, stream, pred, ann, anch, part);
  hipLaunchKernelGGL(yolo_final, dim3(1), dim3(32), 0, stream, part, out);
}